// MultiLabelEmbeddingLayer_3685081940050
// MI455X (gfx1250) — compile-verified
//
#include <hip/hip_runtime.h>
#include <hip/hip_bf16.h>

typedef __attribute__((ext_vector_type(2))) float v2f;
typedef __attribute__((ext_vector_type(8))) float v8f;

// Problem shape (from reference)
constexpr int B = 32, S = 4096, L = 8, V = 100, D = 128;
constexpr int ROWS = B * S;                  // 131072 (b,s) rows
constexpr int TILE_M = 16;                   // WMMA M
constexpr int WAVES_PER_BLOCK = 4;
constexpr int BLOCK = 32 * WAVES_PER_BLOCK;  // 128 threads
// Pair-interleaved table: tbl2[pr][2c+0] = table[2pr][c], tbl2[pr][2c+1] = table[2pr+1][c]
constexpr int PAIR_ROWS   = V / 2;           // 50
constexpr int PAIR_STRIDE = 288;             // dwords; 288%64==32 -> halves hit disjoint banks
constexpr int CNT_STRIDE  = 104;             // bytes/row; 26 dwords -> conflict-free A loads
constexpr int CNT_BYTES_PER_WAVE = TILE_M * CNT_STRIDE;  // 1664 B

__global__ __launch_bounds__(BLOCK)
void MultiLabelEmbeddingLayer_kernel(const int* __restrict__ labels,
                                     const float* __restrict__ table,
                                     float* __restrict__ out) {
  __shared__ float tbl2[PAIR_ROWS * PAIR_STRIDE];                      // 57,600 B
  __shared__ unsigned char cnt[WAVES_PER_BLOCK * CNT_BYTES_PER_WAVE];  //  6,656 B  (total 64,256)

  const int tid  = threadIdx.x;
  const int lane = tid & 31;
  const int wave = tid >> 5;

  // ---- Stage table into LDS, K-pair interleaved, via b128 DS stores ----
  for (int i = tid; i < PAIR_ROWS * (D / 2); i += BLOCK) {  // 3200 col-pair elements
    const int pr = i >> 6;            // 64 column-pairs per pair-row
    const int cc = i & 63;
    const float2 r0 = *(const float2*)&table[(2 * pr)     * D + 2 * cc];
    const float2 r1 = *(const float2*)&table[(2 * pr + 1) * D + 2 * cc];
    float4 v; v.x = r0.x; v.y = r1.x; v.z = r0.y; v.w = r1.y;
    *(float4*)&tbl2[pr * PAIR_STRIDE + 4 * cc] = v;        // 16B aligned
  }

  // ---- Zero this wave's packed-byte count matrix ----
  unsigned int* cw = (unsigned int*)&cnt[wave * CNT_BYTES_PER_WAVE];
  for (int i = lane; i < CNT_BYTES_PER_WAVE / 4; i += 32) cw[i] = 0u;
  __syncthreads();

  // ---- Build 16 x V count matrix: A-operand of the one-hot GEMM ----
  const int tileRow = (blockIdx.x * WAVES_PER_BLOCK + wave) * TILE_M;
  {
    // 16 rows * 8 labels = 128 contiguous int32 per wave; 4 per lane
    const int4 lv = *(const int4*)(labels + tileRow * L + lane * 4);
    const int l4[4] = {lv.x, lv.y, lv.z, lv.w};
#pragma unroll
    for (int j = 0; j < 4; ++j) {
      const int r = (lane * 4 + j) >> 3;            // which of the 16 rows
      const unsigned v = (unsigned)l4[j];           // 0..V-1 (row 0 of table is zero)
      atomicAdd((unsigned int*)&cnt[wave * CNT_BYTES_PER_WAVE + r * CNT_STRIDE + (v & ~3u)],
                1u << ((v & 3u) * 8u));             // counts <= 8, no byte carry
    }
  }
  __syncthreads();

  // ---- out[16 x 128] = counts[16 x V] @ table[V x 128], fp32 WMMA ----
  const int m     = lane & 15;                      // A-matrix row for this lane
  const int khalf = (lane >> 4) << 1;               // 0 (lanes 0-15) or 2 (lanes 16-31)
  const int col   = lane & 15;                      // column within 16-wide N tile
  const unsigned char* cbase = &cnt[wave * CNT_BYTES_PER_WAVE + m * CNT_STRIDE];
  float* outBase = out + (tileRow + ((lane >> 4) << 3)) * D + col;

#pragma unroll
  for (int n = 0; n < D / 16; ++n) {
    v8f acc = {};
    // lane's B base: pair-row offset (khalf/2 = 0 or 1) + interleaved column slot
    const float* bbase = &tbl2[(khalf >> 1) * PAIR_STRIDE + (n * 16 + col) * 2];
    for (int k = 0; k < V; k += 4) {                // 25 exact K-steps (V = 100)
      const int kb = k + khalf;
      // A fragment: counts[m][kb], counts[m][kb+1] (packed bytes -> f32)
      const unsigned short pc = *(const unsigned short*)(cbase + kb);
      v2f a; a.x = (float)(pc & 0xffu); a.y = (float)(pc >> 8);
      // B fragment: one aligned b64 load -> {table[kb][ncol], table[kb+1][ncol]}
      const v2f b = *(const v2f*)(bbase + (k >> 1) * PAIR_STRIDE);
      acc = __builtin_amdgcn_wmma_f32_16x16x4_f32(false, a, false, b,
                                                  (short)0, acc, false, false);
    }
    // C/D layout: VGPR r -> row (r + 8*laneHalf), N = lane&15
#pragma unroll
    for (int r = 0; r < 8; ++r)
      outBase[r * D + n * 16] = acc[r];
  }
}

extern "C" void kernel_launch(void* const* d_in, const int* in_sizes, int n_in,
                              void* d_out, int out_size, void* d_ws, size_t ws_size,
                              hipStream_t stream) {
  (void)in_sizes; (void)n_in; (void)out_size; (void)d_ws; (void)ws_size;
  const int*   labels = (const int*)d_in[0];
  const float* table  = (const float*)d_in[1];
  float*       outp   = (float*)d_out;
  const int nBlocks = ROWS / (TILE_M * WAVES_PER_BLOCK);  // 2048
  MultiLabelEmbeddingLayer_kernel<<<nBlocks, BLOCK, 0, stream>>>(labels, table, outp);
}